// RelationProposalNetwork_36790689858058
// MI455X (gfx1250) — compile-verified
//
#include <hip/hip_runtime.h>

typedef __attribute__((ext_vector_type(2))) float v2f;
typedef __attribute__((ext_vector_type(4))) float v4f;
typedef __attribute__((ext_vector_type(8))) float v8f;

#define INPUT_DIM 64
#define HIDDEN_DIM 128
#define NROWS 8192

// ---------------------------------------------------------------------------
// Kernel 1: fold the two hidden-layer weight matrices through the final
// linear layer:  vs[d] = sum_k Ws[k][d]*Wm[k],  vo[d] = sum_k Wo[k][d]*Wm[128+k]
// plus the scalar constant  C = bs.W1 + bo.W2 + bm.
// Launch: <<<1, 64>>> — negligible cost.
// ---------------------------------------------------------------------------
__global__ void rpn_fold_weights(const float* __restrict__ Ws,
                                 const float* __restrict__ bs,
                                 const float* __restrict__ Wo,
                                 const float* __restrict__ bo,
                                 const float* __restrict__ Wm,
                                 const float* __restrict__ bm,
                                 float* __restrict__ w) {
  // w layout: [0,64) = vs, [64,128) = vo, [128] = C
  const int d = threadIdx.x;  // 0..63
  float as = 0.0f, ao = 0.0f;
  for (int k = 0; k < HIDDEN_DIM; ++k) {
    as = fmaf(Ws[k * INPUT_DIM + d], Wm[k], as);
    ao = fmaf(Wo[k * INPUT_DIM + d], Wm[HIDDEN_DIM + k], ao);
  }
  w[d] = as;
  w[INPUT_DIM + d] = ao;
  if (d == 0) {
    float c = bm[0];
    for (int k = 0; k < HIDDEN_DIM; ++k)
      c += bs[k] * Wm[k] + bo[k] * Wm[HIDDEN_DIM + k];
    w[2 * INPUT_DIM] = c;
  }
}

// ---------------------------------------------------------------------------
// Kernel 2: s[i] = x[i,:].vs ; o[i] = x[i,:].vo via V_WMMA_F32_16X16X4_F32.
// One wave handles a 16-row tile; K=64 marched in 16 steps of 4.
// B matrix: column 0 = vs, column 1 = vo, columns 2..15 zero.
// ISA layouts (CDNA5 ISA §7.12.2), lane = (hi = lane>>4, n = lane&15):
//   A VGPR{0,1} of lane(hi,n)  = A[n][k0+2*hi], A[n][k0+2*hi+1]
//   B VGPR{0,1} of lane(hi,n)  = B[k0+2*hi][n], B[k0+2*hi+1][n]
//   D VGPR r    of lane(hi,n)  = D[8*hi + r][n]
// B-fragment built with unconditional loads (vs/vo are 512 B, cache-hot) and
// v_cndmask selects — no EXEC-divergent guarded loads, no saveexec churn.
// Launch: <<<NROWS/128, 256>>> (8 waves/block * 16 rows/wave).
// ---------------------------------------------------------------------------
__global__ void rpn_scores_wmma(const float* __restrict__ x,
                                const float* __restrict__ w,
                                float* __restrict__ s,
                                float* __restrict__ o) {
#if __has_builtin(__builtin_amdgcn_wmma_f32_16x16x4_f32)
  const int lane = threadIdx.x & 31;
  const int wave = threadIdx.x >> 5;
  const int rowBase = (blockIdx.x * 8 + wave) * 16;
  const int n = lane & 15;
  const int hi = lane >> 4;
  const float* __restrict__ vs = w;
  const float* __restrict__ vo = w + INPUT_DIM;
  const bool isS = (n == 0);
  const bool isO = (n == 1);

  v8f c = {};  // f32 accumulator tile (16x16, 8 VGPRs)
#pragma unroll
  for (int k0 = 0; k0 < INPUT_DIM; k0 += 4) {
    const int ka = k0 + 2 * hi;
    // A fragment: 2 contiguous f32 of this lane's row.
    v2f a = *reinterpret_cast<const v2f*>(
        x + (size_t)(rowBase + n) * INPUT_DIM + ka);
    // B fragment: every lane loads both candidate pairs (uniform-ish, L0-hot),
    // then selects per lane — pure VALU, no divergent control flow.
    v2f bsv = *reinterpret_cast<const v2f*>(vs + ka);
    v2f bov = *reinterpret_cast<const v2f*>(vo + ka);
    v2f b;
    b.x = isS ? bsv.x : (isO ? bov.x : 0.0f);
    b.y = isS ? bsv.y : (isO ? bov.y : 0.0f);
    // 8 args: (neg_a, A, neg_b, B, c_mod, C, reuse_a, reuse_b)
    c = __builtin_amdgcn_wmma_f32_16x16x4_f32(false, a, false, b, (short)0, c,
                                              false, false);
  }

  // D[m][0] = s for row m, D[m][1] = o for row m.
  if (n < 2) {
    float* dst = isS ? s : o;
#pragma unroll
    for (int r = 0; r < 8; ++r)
      dst[rowBase + 8 * hi + r] = c[r];
  }
#else
  // Fallback (no f32 WMMA builtin): one thread per row.
  const int row = blockIdx.x * blockDim.x + threadIdx.x;
  if (row < NROWS) {
    float as = 0.0f, ao = 0.0f;
    for (int k = 0; k < INPUT_DIM; ++k) {
      const float xv = x[(size_t)row * INPUT_DIM + k];
      as = fmaf(xv, w[k], as);
      ao = fmaf(xv, w[INPUT_DIM + k], ao);
    }
    s[row] = as;
    o[row] = ao;
  }
#endif
}

// ---------------------------------------------------------------------------
// Kernel 3: the bandwidth kernel. out[i][j] = sigmoid(s[i] + o[j] + C),
// diagonal forced to sigmoid(0) = 0.5. 256 MiB streaming write — the only
// thing that matters at 23.3 TB/s (~11.5 us floor). 128-bit coalesced
// non-temporal stores (output > 192 MB L2, write-once); o[] is 32 KiB and
// stays hot in cache; s[i]+C is block-uniform (scalar path). Each block
// writes 1024 contiguous floats.
// Launch: <<<NROWS*NROWS/1024, 256>>>.
// ---------------------------------------------------------------------------
__device__ __forceinline__ float fast_sigmoid(float v) {
  return 1.0f / (1.0f + __expf(-v));
}

__global__ void rpn_outer_sigmoid(const float* __restrict__ s,
                                  const float* __restrict__ o,
                                  const float* __restrict__ w,
                                  float* __restrict__ out) {
  const int bid = blockIdx.x;
  const int i = bid >> 3;                                // row (8 blocks/row)
  const int j0 = ((bid & 7) * 256 + threadIdx.x) * 4;    // first of 4 columns
  const float base = s[i] + w[2 * INPUT_DIM];
  const v4f ov = *reinterpret_cast<const v4f*>(o + j0);
  v4f r;
  r.x = fast_sigmoid(base + ov.x);
  r.y = fast_sigmoid(base + ov.y);
  r.z = fast_sigmoid(base + ov.z);
  r.w = fast_sigmoid(base + ov.w);
  if ((unsigned)(i - j0) < 4u) r[i - j0] = 0.5f;         // diagonal
  __builtin_nontemporal_store(
      r, reinterpret_cast<v4f*>(out + (size_t)i * NROWS + j0));
}

// ---------------------------------------------------------------------------
extern "C" void kernel_launch(void* const* d_in, const int* in_sizes, int n_in,
                              void* d_out, int out_size, void* d_ws,
                              size_t ws_size, hipStream_t stream) {
  const float* x  = (const float*)d_in[0];  // (8192, 64)
  const float* Ws = (const float*)d_in[1];  // (128, 64)
  const float* bs = (const float*)d_in[2];  // (128,)
  const float* Wo = (const float*)d_in[3];  // (128, 64)
  const float* bo = (const float*)d_in[4];  // (128,)
  const float* Wm = (const float*)d_in[5];  // (1, 256)
  const float* bm = (const float*)d_in[6];  // (1,)
  float* out = (float*)d_out;               // (8192, 8192)

  float* w = (float*)d_ws;   // [0,128) folded vectors, [128] constant
  float* s = w + 256;        // 8192 floats
  float* o = s + NROWS;      // 8192 floats

  rpn_fold_weights<<<1, 64, 0, stream>>>(Ws, bs, Wo, bo, Wm, bm, w);
  rpn_scores_wmma<<<NROWS / 128, 256, 0, stream>>>(x, w, s, o);
  rpn_outer_sigmoid<<<(NROWS * NROWS) / 1024, 256, 0, stream>>>(s, o, w, out);
}